// Encoder_69243462746519
// MI455X (gfx1250) — compile-verified
//
#include <hip/hip_runtime.h>
#include <hip/hip_bf16.h>

typedef __attribute__((ext_vector_type(2))) float v2f;
typedef __attribute__((ext_vector_type(8))) float v8f;

#define IN_CH 256
#define OUT_CH 512

// ---------------- degree / norm kernels ----------------

__global__ void init_deg_kernel(float* __restrict__ deg, int n) {
    int i = blockIdx.x * blockDim.x + threadIdx.x;
    if (i < n) deg[i] = 1.0f;  // self-loop contributes 1
}

__global__ void count_deg_kernel(const int* __restrict__ dst, float* __restrict__ deg, int n_edges) {
    int e = blockIdx.x * blockDim.x + threadIdx.x;
    if (e < n_edges) atomicAdd(&deg[dst[e]], 1.0f);
}

__global__ void dinv_kernel(float* __restrict__ deg, int n) {
    int i = blockIdx.x * blockDim.x + threadIdx.x;
    if (i < n) deg[i] = rsqrtf(deg[i]);  // deg >= 1 always
}

// xagg[i] = dinv[i]^2 * x[i]   (self-loop term; also zero-initializes the buffer)
__global__ void init_xagg_kernel(const float* __restrict__ x,
                                 const float* __restrict__ dinv,
                                 float* __restrict__ xagg) {
    int row = blockIdx.x;           // one row per block, 256 threads = 256 channels
    int c = threadIdx.x;
    float di = dinv[row];
    size_t idx = (size_t)row * IN_CH + c;
    xagg[idx] = di * di * x[idx];
}

// xagg[dst] += dinv[src]*dinv[dst] * x[src]   (one wave32 per edge, 8 ch/lane, stride-32)
__global__ void edge_scatter_kernel(const float* __restrict__ x,
                                    const int* __restrict__ src,
                                    const int* __restrict__ dst,
                                    const float* __restrict__ dinv,
                                    float* __restrict__ xagg, int n_edges) {
    int wave = threadIdx.x >> 5;
    int lane = threadIdx.x & 31;
    int e = blockIdx.x * (blockDim.x >> 5) + wave;
    if (e >= n_edges) return;
    int s = src[e];
    int d = dst[e];
    float norm = dinv[s] * dinv[d];
    const float* xs = x + (size_t)s * IN_CH;
    float* xd = xagg + (size_t)d * IN_CH;
#pragma unroll
    for (int j = 0; j < 8; ++j) {
        int c = lane + j * 32;                 // coalesced 128B per instruction
        atomicAdd(&xd[c], norm * xs[c]);
    }
}

// ---------------- fused WMMA GEMM + bias + PReLU ----------------
// out[m, n] = prelu( sum_k xagg[m,k] * W[n,k] + bias[n] )
// One wave computes a 16x64 output strip using 4 independent f32 WMMA accumulators.
__global__ void gcn_gemm_wmma_kernel(const float* __restrict__ xagg,   // [N, 256]
                                     const float* __restrict__ W,      // [512, 256]
                                     const float* __restrict__ bias,   // [512]
                                     const float* __restrict__ prelu_a,
                                     float* __restrict__ out,          // [N, 512]
                                     int n_mtiles) {
    const int lane = threadIdx.x & 31;
    const int wave = threadIdx.x >> 5;
    int job = blockIdx.x * 8 + wave;           // 8 waves/block; waves of a block share A
    int mt = job >> 3;                          // 16-row tile index
    int ng = job & 7;                           // 64-col group index
    if (mt >= n_mtiles) return;

    const int m0 = mt * 16;
    const int n0 = ng * 64;
    const int half = lane >> 4;                 // 0: K even pair, 1: K odd pair (+2)
    const int l = lane & 15;

    v8f acc[4] = {v8f{}, v8f{}, v8f{}, v8f{}};

    // A lane base: row (m0+l), element (k + 2*half)
    const float* arow = xagg + (size_t)(m0 + l) * IN_CH + half * 2;
    const float* b0p = W + (size_t)(n0 +  0 + l) * IN_CH + half * 2;
    const float* b1p = W + (size_t)(n0 + 16 + l) * IN_CH + half * 2;
    const float* b2p = W + (size_t)(n0 + 32 + l) * IN_CH + half * 2;
    const float* b3p = W + (size_t)(n0 + 48 + l) * IN_CH + half * 2;

#pragma unroll 4
    for (int k = 0; k < IN_CH; k += 4) {
        v2f a  = *(const v2f*)(arow + k);
        v2f b0 = *(const v2f*)(b0p + k);
        v2f b1 = *(const v2f*)(b1p + k);
        v2f b2 = *(const v2f*)(b2p + k);
        v2f b3 = *(const v2f*)(b3p + k);
        // 8-arg form: (neg_a, A, neg_b, B, c_mod, C, reuse_a, reuse_b)
        acc[0] = __builtin_amdgcn_wmma_f32_16x16x4_f32(false, a, false, b0, (short)0, acc[0], false, false);
        acc[1] = __builtin_amdgcn_wmma_f32_16x16x4_f32(false, a, false, b1, (short)0, acc[1], false, false);
        acc[2] = __builtin_amdgcn_wmma_f32_16x16x4_f32(false, a, false, b2, (short)0, acc[2], false, false);
        acc[3] = __builtin_amdgcn_wmma_f32_16x16x4_f32(false, a, false, b3, (short)0, acc[3], false, false);
    }

    const float pa = prelu_a[0];
#pragma unroll
    for (int nt = 0; nt < 4; ++nt) {
        int col = n0 + nt * 16 + l;
        float bcol = bias[col];
#pragma unroll
        for (int r = 0; r < 8; ++r) {
            // C/D layout: VGPR r -> row m0+r (lanes 0-15), row m0+8+r (lanes 16-31)
            int row = m0 + half * 8 + r;
            float v = acc[nt][r] + bcol;
            out[(size_t)row * OUT_CH + col] = v > 0.0f ? v : pa * v;
        }
    }
}

// ---------------- launcher ----------------

extern "C" void kernel_launch(void* const* d_in, const int* in_sizes, int n_in,
                              void* d_out, int out_size, void* d_ws, size_t ws_size,
                              hipStream_t stream) {
    const float* x       = (const float*)d_in[0];
    const int*   ei      = (const int*)d_in[1];     // [2, E] flat: src row then dst row
    const float* W       = (const float*)d_in[2];   // [512, 256]
    const float* bias    = (const float*)d_in[3];
    const float* prelu_a = (const float*)d_in[4];
    float* out = (float*)d_out;

    const int N = in_sizes[0] / IN_CH;     // 50000
    const int E = in_sizes[1] / 2;         // 400000
    const int* src = ei;
    const int* dst = ei + E;

    // workspace layout: dinv [N floats, 256B aligned pad] | xagg [N*256 floats]
    float* dinv = (float*)d_ws;
    size_t dinv_elems = ((size_t)N + 63) & ~(size_t)63;
    float* xagg = dinv + dinv_elems;

    // 1) degree (with self-loop) -> dinv = rsqrt(deg)
    init_deg_kernel<<<(N + 255) / 256, 256, 0, stream>>>(dinv, N);
    count_deg_kernel<<<(E + 255) / 256, 256, 0, stream>>>(dst, dinv, E);
    dinv_kernel<<<(N + 255) / 256, 256, 0, stream>>>(dinv, N);

    // 2) input-space aggregation: xagg = (D^-1/2 A_hat D^-1/2) X
    init_xagg_kernel<<<N, 256, 0, stream>>>(x, dinv, xagg);
    edge_scatter_kernel<<<(E + 7) / 8, 256, 0, stream>>>(x, src, dst, dinv, xagg, E);

    // 3) fused WMMA GEMM (xagg @ W^T) + bias + PReLU
    int n_mtiles = N / 16;                         // 50000/16 = 3125 exact
    int jobs = n_mtiles * 8;                       // 8 n-groups of 64 cols
    gcn_gemm_wmma_kernel<<<(jobs + 7) / 8, 256, 0, stream>>>(xagg, W, bias, prelu_a, out, n_mtiles);
}